// GraphAttentionLayerV2_71442486001856
// MI455X (gfx1250) — compile-verified
//
#include <hip/hip_runtime.h>

// ---------------------------------------------------------------------------
// GATv2 layer for MI455X (gfx1250, wave32).
//   Wh1 = h @ W[:128], Wh2 = h @ W[128:]           -> fp32 WMMA 16x16x4
//   e[i,k] = sum_f a[f]*lrelu(Wh1[i,f]+Wh2[k,f])   -> VALU (nonlinear reduce)
//   att = softmax(mask(e))                          -> LDS row softmax
//   out = elu(att @ Wh1)                            -> fp32 WMMA 16x16x4, K=2048
// Dominant traffic: adj (16MB) + att (16MB write + 16MB read, L2-resident).
// ---------------------------------------------------------------------------

#define N_NODES 2048
#define IN_F    128
#define OUT_F   32
#define ALPHA   0.2f
#define NEG_INF (-9.0e15f)

typedef __attribute__((ext_vector_type(2))) float v2f;
typedef __attribute__((ext_vector_type(8))) float v8f;

__device__ __forceinline__ v8f wmma_f32_16x16x4(v2f a, v2f b, v8f c) {
  // 8 args: (neg_a, A, neg_b, B, c_mod, C, reuse_a, reuse_b)
  return __builtin_amdgcn_wmma_f32_16x16x4_f32(
      false, a, false, b, (short)0, c, false, false);
}

// ---------------------------------------------------------------------------
// Kernel 1: Wh1/Wh2 GEMM.  One wave (32 lanes) per 16x16 output tile.
// blockIdx.x in [0,4): bit1 selects W half (Wh1/Wh2), bit0 selects col tile.
// blockIdx.y: row tile (2048/16 = 128).
// ---------------------------------------------------------------------------
__global__ void __launch_bounds__(32)
wh_gemm_kernel(const float* __restrict__ h, const float* __restrict__ W,
               float* __restrict__ Wh1, float* __restrict__ Wh2) {
  const int lane = threadIdx.x;            // 0..31
  const int c    = blockIdx.x;             // 0..3
  const int i0   = blockIdx.y * 16;
  const int half = c >> 1;
  const int col0 = (c & 1) * 16;
  const int wrow0 = half * IN_F;
  float* __restrict__ out = half ? Wh2 : Wh1;

  // A fragment (16x4 f32): lane&15 = M row, (lane>>4)*2 = K offset pair.
  // B fragment (4x16 f32): lane&15 = N col, (lane>>4)*2 = K offset pair.
  const int m  = lane & 15;
  const int n  = lane & 15;
  const int kb = (lane >> 4) * 2;

  const float* __restrict__ hrow = h + (size_t)(i0 + m) * IN_F;

  v8f acc = {};
  for (int k0 = 0; k0 < IN_F; k0 += 4) {
    v2f A, B;
    A.x = hrow[k0 + kb];
    A.y = hrow[k0 + kb + 1];
    B.x = W[(size_t)(wrow0 + k0 + kb)     * OUT_F + col0 + n];
    B.y = W[(size_t)(wrow0 + k0 + kb + 1) * OUT_F + col0 + n];
    acc = wmma_f32_16x16x4(A, B, acc);
  }

  // C/D layout: VGPR r -> M=r (lanes 0-15) / M=8+r (lanes 16-31), N=lane&15.
  const int mh = (lane >> 4) * 8;
#pragma unroll
  for (int r = 0; r < 8; ++r)
    out[(size_t)(i0 + mh + r) * OUT_F + col0 + n] = acc[r];
}

// ---------------------------------------------------------------------------
// Kernel 2: per-row scores + mask + softmax -> att row in workspace.
// One 256-thread block (8 waves) per row i. e-row staged in LDS (8KB).
// ---------------------------------------------------------------------------
__global__ void __launch_bounds__(256)
attn_softmax_kernel(const float* __restrict__ Wh1, const float* __restrict__ Wh2,
                    const int* __restrict__ adj, const float* __restrict__ a,
                    float* __restrict__ att) {
  __shared__ float e_row[N_NODES];
  __shared__ float red[256];
  __shared__ float wh1_s[OUT_F];
  __shared__ float a_s[OUT_F];

  const int i = blockIdx.x;
  const int t = threadIdx.x;

  if (t < OUT_F) {
    wh1_s[t] = Wh1[(size_t)i * OUT_F + t];
    a_s[t]   = a[t];
  }
  __syncthreads();

  float lmax = -3.0e38f;
  for (int k = t; k < N_NODES; k += 256) {
    float s;
    if (adj[(size_t)i * N_NODES + k] > 0) {
      const float4* __restrict__ w2 =
          (const float4*)(Wh2 + (size_t)k * OUT_F);   // 128B-aligned rows
      s = 0.0f;
#pragma unroll
      for (int q = 0; q < OUT_F / 4; ++q) {
        float4 v = w2[q];
        float x0 = wh1_s[4 * q + 0] + v.x; s += a_s[4 * q + 0] * (x0 > 0.f ? x0 : ALPHA * x0);
        float x1 = wh1_s[4 * q + 1] + v.y; s += a_s[4 * q + 1] * (x1 > 0.f ? x1 : ALPHA * x1);
        float x2 = wh1_s[4 * q + 2] + v.z; s += a_s[4 * q + 2] * (x2 > 0.f ? x2 : ALPHA * x2);
        float x3 = wh1_s[4 * q + 3] + v.w; s += a_s[4 * q + 3] * (x3 > 0.f ? x3 : ALPHA * x3);
      }
    } else {
      s = NEG_INF;
    }
    e_row[k] = s;
    lmax = fmaxf(lmax, s);
  }

  // block max
  red[t] = lmax;
  __syncthreads();
  for (int s2 = 128; s2 > 0; s2 >>= 1) {
    if (t < s2) red[t] = fmaxf(red[t], red[t + s2]);
    __syncthreads();
  }
  const float m = red[0];
  __syncthreads();

  // exp + block sum
  float lsum = 0.0f;
  for (int k = t; k < N_NODES; k += 256) {
    float p = expf(e_row[k] - m);   // masked entries -> exp(-huge) == 0
    e_row[k] = p;
    lsum += p;
  }
  red[t] = lsum;
  __syncthreads();
  for (int s2 = 128; s2 > 0; s2 >>= 1) {
    if (t < s2) red[t] += red[t + s2];
    __syncthreads();
  }
  const float invZ = 1.0f / red[0];

  for (int k = t; k < N_NODES; k += 256)
    att[(size_t)i * N_NODES + k] = e_row[k] * invZ;
}

// ---------------------------------------------------------------------------
// Kernel 3: out = elu(att @ Wh1).  One wave per 16x16 tile; K = 2048
// -> 512 v_wmma_f32_16x16x4_f32 per tile, ELU fused on the accumulator.
// ---------------------------------------------------------------------------
__global__ void __launch_bounds__(32)
aggregate_wmma_kernel(const float* __restrict__ att, const float* __restrict__ Wh1,
                      float* __restrict__ out) {
  const int lane = threadIdx.x;
  const int col0 = blockIdx.x * 16;        // 0 or 16
  const int i0   = blockIdx.y * 16;

  const int m  = lane & 15;
  const int n  = lane & 15;
  const int kb = (lane >> 4) * 2;

  const float* __restrict__ arow = att + (size_t)(i0 + m) * N_NODES;

  v8f acc = {};
  for (int k0 = 0; k0 < N_NODES; k0 += 4) {
    v2f A, B;
    A.x = arow[k0 + kb];
    A.y = arow[k0 + kb + 1];
    B.x = Wh1[(size_t)(k0 + kb)     * OUT_F + col0 + n];
    B.y = Wh1[(size_t)(k0 + kb + 1) * OUT_F + col0 + n];
    acc = wmma_f32_16x16x4(A, B, acc);
  }

  const int mh = (lane >> 4) * 8;
#pragma unroll
  for (int r = 0; r < 8; ++r) {
    float v = acc[r];
    v = v > 0.0f ? v : (expf(v) - 1.0f);   // ELU (alpha = 1)
    out[(size_t)(i0 + mh + r) * OUT_F + col0 + n] = v;
  }
}

// ---------------------------------------------------------------------------
// Fallback (only if workspace can't hold the 16MB att matrix): fused
// scores + softmax + VALU aggregation per row. Still uses WMMA kernel 1.
// ---------------------------------------------------------------------------
__global__ void __launch_bounds__(256)
attn_fused_kernel(const float* __restrict__ Wh1, const float* __restrict__ Wh2,
                  const int* __restrict__ adj, const float* __restrict__ a,
                  float* __restrict__ out) {
  __shared__ float e_row[N_NODES];
  __shared__ float red[256];
  __shared__ float wh1_s[OUT_F];
  __shared__ float a_s[OUT_F];
  __shared__ float agg[OUT_F * 256];   // 32KB (LDS is 320KB/WGP on CDNA5)

  const int i = blockIdx.x;
  const int t = threadIdx.x;

  if (t < OUT_F) {
    wh1_s[t] = Wh1[(size_t)i * OUT_F + t];
    a_s[t]   = a[t];
  }
  __syncthreads();

  float lmax = -3.0e38f;
  for (int k = t; k < N_NODES; k += 256) {
    float s;
    if (adj[(size_t)i * N_NODES + k] > 0) {
      const float4* __restrict__ w2 = (const float4*)(Wh2 + (size_t)k * OUT_F);
      s = 0.0f;
#pragma unroll
      for (int q = 0; q < OUT_F / 4; ++q) {
        float4 v = w2[q];
        float x0 = wh1_s[4 * q + 0] + v.x; s += a_s[4 * q + 0] * (x0 > 0.f ? x0 : ALPHA * x0);
        float x1 = wh1_s[4 * q + 1] + v.y; s += a_s[4 * q + 1] * (x1 > 0.f ? x1 : ALPHA * x1);
        float x2 = wh1_s[4 * q + 2] + v.z; s += a_s[4 * q + 2] * (x2 > 0.f ? x2 : ALPHA * x2);
        float x3 = wh1_s[4 * q + 3] + v.w; s += a_s[4 * q + 3] * (x3 > 0.f ? x3 : ALPHA * x3);
      }
    } else {
      s = NEG_INF;
    }
    e_row[k] = s;
    lmax = fmaxf(lmax, s);
  }

  red[t] = lmax;
  __syncthreads();
  for (int s2 = 128; s2 > 0; s2 >>= 1) {
    if (t < s2) red[t] = fmaxf(red[t], red[t + s2]);
    __syncthreads();
  }
  const float m = red[0];
  __syncthreads();

  float lsum = 0.0f;
  for (int k = t; k < N_NODES; k += 256) {
    float p = expf(e_row[k] - m);
    e_row[k] = p;
    lsum += p;
  }
  red[t] = lsum;
  __syncthreads();
  for (int s2 = 128; s2 > 0; s2 >>= 1) {
    if (t < s2) red[t] += red[t + s2];
    __syncthreads();
  }
  const float invZ = 1.0f / red[0];

  float partial[OUT_F];
#pragma unroll
  for (int f = 0; f < OUT_F; ++f) partial[f] = 0.0f;

  for (int k = t; k < N_NODES; k += 256) {
    const float p = e_row[k] * invZ;
    const float4* __restrict__ w1 = (const float4*)(Wh1 + (size_t)k * OUT_F);
#pragma unroll
    for (int q = 0; q < OUT_F / 4; ++q) {
      float4 v = w1[q];
      partial[4 * q + 0] += p * v.x;
      partial[4 * q + 1] += p * v.y;
      partial[4 * q + 2] += p * v.z;
      partial[4 * q + 3] += p * v.w;
    }
  }
#pragma unroll
  for (int f = 0; f < OUT_F; ++f) agg[f * 256 + t] = partial[f];
  __syncthreads();

  if (t < OUT_F) {
    float s = 0.0f;
    for (int j = 0; j < 256; ++j) s += agg[t * 256 + j];
    s = s > 0.0f ? s : (expf(s) - 1.0f);
    out[(size_t)i * OUT_F + t] = s;
  }
}

// ---------------------------------------------------------------------------
extern "C" void kernel_launch(void* const* d_in, const int* in_sizes, int n_in,
                              void* d_out, int out_size, void* d_ws, size_t ws_size,
                              hipStream_t stream) {
  (void)in_sizes; (void)n_in; (void)out_size;
  const float* h   = (const float*)d_in[0];   // [2048,128] f32
  const int*   adj = (const int*)  d_in[1];   // [2048,2048] i32
  const float* W   = (const float*)d_in[2];   // [256,32] f32
  const float* a   = (const float*)d_in[3];   // [32] f32
  float* out = (float*)d_out;                 // [2048,32] f32
  float* ws  = (float*)d_ws;

  float* Wh1 = ws;                                    // 2048*32 f32 (256KB)
  float* Wh2 = ws + (size_t)N_NODES * OUT_F;          // 2048*32 f32 (256KB)
  float* att = ws + (size_t)2 * N_NODES * OUT_F;      // 2048*2048 f32 (16MB)

  // Stage 1: Wh1/Wh2 via fp32 WMMA.
  dim3 g1(4, N_NODES / 16);
  wh_gemm_kernel<<<g1, 32, 0, stream>>>(h, W, Wh1, Wh2);

  const size_t need =
      ((size_t)2 * N_NODES * OUT_F + (size_t)N_NODES * N_NODES) * sizeof(float);
  if (ws_size >= need) {
    // Stage 2: masked scores + row softmax -> att (L2-resident, 16MB).
    attn_softmax_kernel<<<N_NODES, 256, 0, stream>>>(Wh1, Wh2, adj, a, att);
    // Stage 3: elu(att @ Wh1) via fp32 WMMA, K=2048.
    dim3 g3(OUT_F / 16, N_NODES / 16);
    aggregate_wmma_kernel<<<g3, 32, 0, stream>>>(att, Wh1, out);
  } else {
    attn_fused_kernel<<<N_NODES, 256, 0, stream>>>(Wh1, Wh2, adj, a, out);
  }
}